// SVDHead_29523605193338
// MI455X (gfx1250) — compile-verified
//
#include <hip/hip_runtime.h>
#include <stdint.h>

// ---------------------------------------------------------------------------
// SVD head (weighted Kabsch) for B=1024, N=4096.
// Stage 1: HBM-bandwidth-bound streaming reduction of 22 sufficient statistics
//          per batch. CDNA5 paths used:
//            - TDM tensor_load_to_lds (TENSORcnt) for the strided 3x128 tiles
//              of src / src_corr (one scalar DMA op per tile per wave)
//            - global_load_async_to_lds_b128 (ASYNCcnt) for the flat weights
//            - V_WMMA_F32_16X16X4_F32 for the cross-wave combine:
//              totals = ones(1x8) x red(8x22), two chained K=4 WMMAs per
//              16-stat chunk (A = stats-per-lane in native 16x4 layout).
//          Main loop stays VALU FMA: AI ~1.3 FLOP/byte => memory-bound; f32
//          WMMA in the inner loop would need an LDS lane-transpose costing
//          more DS ops than the FMAs it replaces.
// Stage 2: per-batch 3x3 SVD (Jacobi on H^T H) -> R, t.
// ---------------------------------------------------------------------------

#define B_BATCH 1024
#define NPTS    4096
#define TILE    128                       // n-elements per wave per tile
#define NWAVE   8
#define NSTREAM 7                         // sx,sy,sz,cx,cy,cz,w
#define STREAM_BYTES (TILE * 4)           // 512
#define BUF_BYTES    (NSTREAM * STREAM_BYTES)   // 3584
#define WAVE_BYTES   (2 * BUF_BYTES)            // 7168 (double buffer)
#define STAGE_BYTES  (NWAVE * WAVE_BYTES)       // 57344
#define RED_STRIDE   32                         // padded stat stride per wave
#define RED_FLOATS   (NWAVE * RED_STRIDE)       // 256
#define TOT_FLOATS   32

typedef unsigned int v4u __attribute__((ext_vector_type(4)));
typedef int          v8i __attribute__((ext_vector_type(8)));
typedef int          v4i __attribute__((ext_vector_type(4)));
typedef float        v2f __attribute__((ext_vector_type(2)));
typedef float        v8f __attribute__((ext_vector_type(8)));

#define TENSOR_WAIT_2() asm volatile("s_wait_tensorcnt 0x2" ::: "memory")
#define TENSOR_WAIT_0() asm volatile("s_wait_tensorcnt 0x0" ::: "memory")
#define ASYNC_WAIT_1()  asm volatile("s_wait_asynccnt 0x1" ::: "memory")
#define ASYNC_WAIT_0()  asm volatile("s_wait_asynccnt 0x0" ::: "memory")

// GVS-mode per-lane async copy: SGPR 64-bit base + 32-bit byte offset -> LDS.
__device__ __forceinline__ void async_b128(unsigned int lds_addr,
                                           const float* base,
                                           unsigned int byte_off) {
  asm volatile("global_load_async_to_lds_b128 %0, %1, %2"
               :: "v"(lds_addr), "v"(byte_off), "s"(base)
               : "memory");
}

// TDM: load a (rows x TILE) f32 tile with row pitch NPTS elements into LDS,
// rows landing contiguously at 512B stride (== STREAM_BYTES). ISA §8.3/§8.4.
__device__ __forceinline__ void tdm_load_tile(unsigned int lds_addr,
                                              const float* gptr,
                                              unsigned int rows) {
  unsigned long long ga = (unsigned long long)(size_t)gptr;   // byte address
  v4u g0;
  g0.x = 1u;                                                   // count=1, user desc
  g0.y = lds_addr;                                             // lds_addr [63:32]
  g0.z = (unsigned int)ga;                                     // global_addr lo
  g0.w = (unsigned int)((ga >> 32) & 0x01FFFFFFull)            // global_addr[56:32]
       | 0x80000000u;                                          // type=2 ("image")
  v8i g1;
  g1[0] = (int)(2u << 16);                                     // data_size=4B
  g1[1] = (int)(((unsigned)NPTS & 0xFFFFu) << 16);             // tensor_dim0 lo16
  g1[2] = (int)(((unsigned)NPTS >> 16) | (rows << 16));        // dim0 hi16 | tensor_dim1
  g1[3] = (int)((unsigned)TILE << 16);                         // tile_dim0
  g1[4] = (int)rows;                                           // tile_dim1 (tile_dim2=0)
  g1[5] = (int)NPTS;                                           // tensor_dim0_stride lo32
  g1[6] = 0;                                                   // stride hi | dim1_stride lo
  g1[7] = 0;
  v4i z4 = {0, 0, 0, 0};                                       // <=2D: groups 2/3 unused
  v8i z8 = {0, 0, 0, 0, 0, 0, 0, 0};
  __builtin_amdgcn_tensor_load_to_lds(g0, g1, z4, z4, z8, 0);
}

__device__ __forceinline__ void accum(float* a, float sx, float sy, float sz,
                                      float cx, float cy, float cz, float w) {
  float wx = w * sx, wy = w * sy, wz = w * sz;
  a[0]  = fmaf(wx, cx, a[0]);  a[1]  = fmaf(wx, cy, a[1]);  a[2]  = fmaf(wx, cz, a[2]);
  a[3]  = fmaf(wy, cx, a[3]);  a[4]  = fmaf(wy, cy, a[4]);  a[5]  = fmaf(wy, cz, a[5]);
  a[6]  = fmaf(wz, cx, a[6]);  a[7]  = fmaf(wz, cy, a[7]);  a[8]  = fmaf(wz, cz, a[8]);
  a[9]  += wx;  a[10] += wy;  a[11] += wz;                        // p = sum w*s
  a[12] = fmaf(w, cx, a[12]); a[13] = fmaf(w, cy, a[13]); a[14] = fmaf(w, cz, a[14]); // q
  a[15] += w;                                                     // W
  a[16] += sx; a[17] += sy; a[18] += sz;                          // sum s
  a[19] += cx; a[20] += cy; a[21] += cz;                          // sum c
}

__global__ __launch_bounds__(256) void moments_kernel(const float* __restrict__ src,
                                                      const float* __restrict__ corr,
                                                      const float* __restrict__ wts,
                                                      float* __restrict__ ws) {
  __shared__ __align__(16) unsigned char smem[STAGE_BYTES + (RED_FLOATS + TOT_FLOATS) * 4];

  const int b    = blockIdx.x;
  const int lane = (int)(threadIdx.x & 31);
  const int wave = __builtin_amdgcn_readfirstlane((int)(threadIdx.x >> 5)); // SGPR-uniform

  const float* srcB  = src  + (size_t)b * 3 * NPTS;
  const float* corrB = corr + (size_t)b * 3 * NPTS;
  const float* wp    = wts  + (size_t)b * NPTS;

  const unsigned int l16      = (unsigned int)lane << 4;
  const unsigned int smemBase = (unsigned int)(size_t)(&smem[0]); // low 32b = LDS offset
  const unsigned int waveBase = smemBase + (unsigned int)wave * WAVE_BYTES;
  const unsigned int n0       = (unsigned int)wave * (NPTS / NWAVE); // 512 per wave

  auto issue_tile = [&](int t) {
    const unsigned int ldsBuf = waveBase + (unsigned int)(t & 1) * BUF_BYTES;
    const unsigned int nt     = n0 + (unsigned int)t * TILE;
    tdm_load_tile(ldsBuf + 0u * STREAM_BYTES, srcB  + nt, 3u);  // streams 0..2
    tdm_load_tile(ldsBuf + 3u * STREAM_BYTES, corrB + nt, 3u);  // streams 3..5
    async_b128(ldsBuf + 6u * STREAM_BYTES + l16, wp, nt * 4u + l16); // stream 6
  };

  issue_tile(0);                      // prologue -> buffer 0

  float acc[22];
#pragma unroll
  for (int j = 0; j < 22; ++j) acc[j] = 0.0f;

#pragma unroll
  for (int t = 0; t < 4; ++t) {       // 4 tiles of 128 per wave
    if (t < 3) {
      issue_tile(t + 1);              // next tile in flight (2 TDM + 1 async)
      TENSOR_WAIT_2();                // current tile's 2 TDM ops complete
      ASYNC_WAIT_1();                 // current tile's weight copy complete
    } else {
      TENSOR_WAIT_0();
      ASYNC_WAIT_0();
    }
    const unsigned char* bp = &smem[(size_t)wave * WAVE_BYTES + (size_t)(t & 1) * BUF_BYTES];
    const float4 vsx = *reinterpret_cast<const float4*>(bp + 0 * STREAM_BYTES + l16);
    const float4 vsy = *reinterpret_cast<const float4*>(bp + 1 * STREAM_BYTES + l16);
    const float4 vsz = *reinterpret_cast<const float4*>(bp + 2 * STREAM_BYTES + l16);
    const float4 vcx = *reinterpret_cast<const float4*>(bp + 3 * STREAM_BYTES + l16);
    const float4 vcy = *reinterpret_cast<const float4*>(bp + 4 * STREAM_BYTES + l16);
    const float4 vcz = *reinterpret_cast<const float4*>(bp + 5 * STREAM_BYTES + l16);
    const float4 vw  = *reinterpret_cast<const float4*>(bp + 6 * STREAM_BYTES + l16);
    accum(acc, vsx.x, vsy.x, vsz.x, vcx.x, vcy.x, vcz.x, vw.x);
    accum(acc, vsx.y, vsy.y, vsz.y, vcx.y, vcy.y, vcz.y, vw.y);
    accum(acc, vsx.z, vsy.z, vsz.z, vcx.z, vcy.z, vcz.z, vw.z);
    accum(acc, vsx.w, vsy.w, vsz.w, vcx.w, vcy.w, vcz.w, vw.w);
  }

  // wave32 shuffle reduction -> red[wave][stat]
#pragma unroll
  for (int j = 0; j < 22; ++j) {
    float v = acc[j];
#pragma unroll
    for (int o = 16; o > 0; o >>= 1) v += __shfl_down(v, o, 32);
    acc[j] = v;
  }
  float* red = reinterpret_cast<float*>(&smem[STAGE_BYTES]);
  float* tot = red + RED_FLOATS;
  if (lane == 0) {
#pragma unroll
    for (int j = 0; j < 22; ++j) red[wave * RED_STRIDE + j] = acc[j];
  }
  __syncthreads();

  // Cross-wave combine on the matrix pipe: totals = ones(1x8) x red(8x22).
  // Two chained V_WMMA_F32_16X16X4_F32 (K=4 waves each) per 16-stat chunk.
  // A (16x4 f32, 2 VGPRs): lane 0-15 = stat row, K = 2*laneHalf + vgpr.
  // C (16x16 f32): VGPR r = stat row r (lanes 0-15) / r+8 (lanes 16-31).
  if (wave == 0) {
    const int row   = lane & 15;
    const int khalf = lane >> 4;
#pragma unroll
    for (int chunk = 0; chunk < 2; ++chunk) {   // stats [0,16) and [16,22) (+pad)
      const int cbase = chunk * 16;
      v2f aLo, aHi, ones;
      aLo[0] = red[(2 * khalf + 0) * RED_STRIDE + cbase + row];   // waves 0..3
      aLo[1] = red[(2 * khalf + 1) * RED_STRIDE + cbase + row];
      aHi[0] = red[(2 * khalf + 4) * RED_STRIDE + cbase + row];   // waves 4..7
      aHi[1] = red[(2 * khalf + 5) * RED_STRIDE + cbase + row];
      ones[0] = 1.0f; ones[1] = 1.0f;
      v8f c = {0.0f, 0.0f, 0.0f, 0.0f, 0.0f, 0.0f, 0.0f, 0.0f};
      c = __builtin_amdgcn_wmma_f32_16x16x4_f32(false, aLo, false, ones,
                                                (short)0, c, false, false);
      c = __builtin_amdgcn_wmma_f32_16x16x4_f32(false, aHi, false, ones,
                                                (short)0, c, false, false);
      if (row == 0) {                            // lanes 0 and 16: column N=0
        const int sbase = cbase + khalf * 8;
#pragma unroll
        for (int r = 0; r < 8; ++r) tot[sbase + r] = c[r];
      }
    }
    // finalize (same wave: LDS ops are in program order, no barrier needed)
    if (lane == 0) {
      const float invN = 1.0f / (float)NPTS;
      float m[3]  = {tot[16] * invN, tot[17] * invN, tot[18] * invN}; // mean src
      float nn[3] = {tot[19] * invN, tot[20] * invN, tot[21] * invN}; // mean corr
      float W = tot[15];
      float* o = ws + (size_t)b * 16;
#pragma unroll
      for (int i = 0; i < 3; ++i)
#pragma unroll
        for (int j = 0; j < 3; ++j)
          o[i * 3 + j] = tot[i * 3 + j] - m[i] * tot[12 + j]
                       - tot[9 + i] * nn[j] + W * m[i] * nn[j];
#pragma unroll
      for (int i = 0; i < 3; ++i) { o[9 + i] = tot[9 + i]; o[12 + i] = tot[12 + i]; }
    }
  }
}

// ---------------------------------------------------------------------------
// Stage 2: per-batch 3x3 SVD via Jacobi on A = H^T H, then R = V' U^T, t = q - R p
// ---------------------------------------------------------------------------

__device__ __forceinline__ void jrot(float A[3][3], float V[3][3], int p, int q) {
  float apq = A[p][q];
  if (fabsf(apq) < 1e-30f) return;
  float theta = (A[q][q] - A[p][p]) / (2.0f * apq);
  float t = copysignf(1.0f, theta) / (fabsf(theta) + sqrtf(theta * theta + 1.0f));
  float c = 1.0f / sqrtf(t * t + 1.0f);
  float s = t * c;
#pragma unroll
  for (int k = 0; k < 3; ++k) {            // A <- A*G
    float akp = A[k][p], akq = A[k][q];
    A[k][p] = c * akp - s * akq;
    A[k][q] = s * akp + c * akq;
  }
#pragma unroll
  for (int k = 0; k < 3; ++k) {            // A <- G^T*A
    float apk = A[p][k], aqk = A[q][k];
    A[p][k] = c * apk - s * aqk;
    A[q][k] = s * apk + c * aqk;
  }
#pragma unroll
  for (int k = 0; k < 3; ++k) {            // V <- V*G
    float vkp = V[k][p], vkq = V[k][q];
    V[k][p] = c * vkp - s * vkq;
    V[k][q] = s * vkp + c * vkq;
  }
}

__global__ __launch_bounds__(128) void svd_kernel(const float* __restrict__ ws,
                                                  float* __restrict__ out) {
  const int b = blockIdx.x * blockDim.x + threadIdx.x;
  if (b >= B_BATCH) return;
  const float* d = ws + (size_t)b * 16;

  float H[3][3], p[3], q[3];
#pragma unroll
  for (int i = 0; i < 3; ++i)
#pragma unroll
    for (int j = 0; j < 3; ++j) H[i][j] = d[i * 3 + j];
  p[0] = d[9];  p[1] = d[10]; p[2] = d[11];
  q[0] = d[12]; q[1] = d[13]; q[2] = d[14];

  float A[3][3];
#pragma unroll
  for (int i = 0; i < 3; ++i)
#pragma unroll
    for (int j = 0; j < 3; ++j)
      A[i][j] = H[0][i] * H[0][j] + H[1][i] * H[1][j] + H[2][i] * H[2][j];

  float V[3][3] = {{1, 0, 0}, {0, 1, 0}, {0, 0, 1}};
#pragma unroll
  for (int sweep = 0; sweep < 6; ++sweep) {
    jrot(A, V, 0, 1); jrot(A, V, 0, 2); jrot(A, V, 1, 2);
  }

  float lam0 = A[0][0], lam1 = A[1][1], lam2 = A[2][2];
#define SWAPCOL(i, j, li, lj)                                              \
  { float tl = li; li = lj; lj = tl;                                       \
    float t0 = V[0][i]; V[0][i] = V[0][j]; V[0][j] = t0;                   \
    float t1 = V[1][i]; V[1][i] = V[1][j]; V[1][j] = t1;                   \
    float t2 = V[2][i]; V[2][i] = V[2][j]; V[2][j] = t2; }
  if (lam0 < lam1) SWAPCOL(0, 1, lam0, lam1);
  if (lam0 < lam2) SWAPCOL(0, 2, lam0, lam2);
  if (lam1 < lam2) SWAPCOL(1, 2, lam1, lam2);
#undef SWAPCOL

  float u0[3], u1[3], u2[3];
#pragma unroll
  for (int i = 0; i < 3; ++i)
    u0[i] = H[i][0] * V[0][0] + H[i][1] * V[1][0] + H[i][2] * V[2][0];
  float n0 = sqrtf(u0[0] * u0[0] + u0[1] * u0[1] + u0[2] * u0[2]);
  float i0 = 1.0f / fmaxf(n0, 1e-20f);
  u0[0] *= i0; u0[1] *= i0; u0[2] *= i0;
#pragma unroll
  for (int i = 0; i < 3; ++i)
    u1[i] = H[i][0] * V[0][1] + H[i][1] * V[1][1] + H[i][2] * V[2][1];
  float dp = u1[0] * u0[0] + u1[1] * u0[1] + u1[2] * u0[2];
  u1[0] -= dp * u0[0]; u1[1] -= dp * u0[1]; u1[2] -= dp * u0[2];
  float n1 = sqrtf(u1[0] * u1[0] + u1[1] * u1[1] + u1[2] * u1[2]);
  float i1 = 1.0f / fmaxf(n1, 1e-20f);
  u1[0] *= i1; u1[1] *= i1; u1[2] *= i1;
  u2[0] = u0[1] * u1[2] - u0[2] * u1[1];
  u2[1] = u0[2] * u1[0] - u0[0] * u1[2];
  u2[2] = u0[0] * u1[1] - u0[1] * u1[0];

  float R[3][3];
#pragma unroll
  for (int i = 0; i < 3; ++i)
#pragma unroll
    for (int j = 0; j < 3; ++j)
      R[i][j] = V[i][0] * u0[j] + V[i][1] * u1[j] + V[i][2] * u2[j];
  float det = R[0][0] * (R[1][1] * R[2][2] - R[1][2] * R[2][1])
            - R[0][1] * (R[1][0] * R[2][2] - R[1][2] * R[2][0])
            + R[0][2] * (R[1][0] * R[2][1] - R[1][1] * R[2][0]);
  float flipm1 = (det < 0.0f) ? -2.0f : 0.0f;   // (flip - 1)
#pragma unroll
  for (int i = 0; i < 3; ++i)
#pragma unroll
    for (int j = 0; j < 3; ++j)
      R[i][j] += flipm1 * V[i][2] * u2[j];

  float* outR = out + (size_t)b * 9;
#pragma unroll
  for (int i = 0; i < 3; ++i)
#pragma unroll
    for (int j = 0; j < 3; ++j) outR[i * 3 + j] = R[i][j];
  float* outT = out + (size_t)B_BATCH * 9 + (size_t)b * 3;
#pragma unroll
  for (int i = 0; i < 3; ++i)
    outT[i] = q[i] - (R[i][0] * p[0] + R[i][1] * p[1] + R[i][2] * p[2]);
}

extern "C" void kernel_launch(void* const* d_in, const int* in_sizes, int n_in,
                              void* d_out, int out_size, void* d_ws, size_t ws_size,
                              hipStream_t stream) {
  (void)in_sizes; (void)n_in; (void)out_size; (void)ws_size;
  const float* src  = (const float*)d_in[0];
  const float* corr = (const float*)d_in[1];
  const float* wts  = (const float*)d_in[2];
  float* ws  = (float*)d_ws;
  float* out = (float*)d_out;

  moments_kernel<<<B_BATCH, 256, 0, stream>>>(src, corr, wts, ws);
  svd_kernel<<<(B_BATCH + 127) / 128, 128, 0, stream>>>(ws, out);
}